// CtcLossLayer_37005438222686
// MI455X (gfx1250) — compile-verified
//
#include <hip/hip_runtime.h>
#include <math.h>

typedef __attribute__((ext_vector_type(2))) float v2f;
typedef __attribute__((ext_vector_type(4))) float v4f;
typedef __attribute__((ext_vector_type(8))) float v8f;

#define B_ 256
#define T_ 256
#define C_ 512
#define L_ 64
#define S_ 129          // 2L+1
#define NEGV (-1e30f)

// ---------------------------------------------------------------------------
// Kernel 1: lse[b,t] = logsumexp_c logits[b,t,:]  (row = b*T + t, 512 wide)
// 16 rows per wave; sum-of-exp done with V_WMMA_F32_16X16X4_F32 vs ones.
// ---------------------------------------------------------------------------
__global__ __launch_bounds__(128) void lse_kernel(const float* __restrict__ logits,
                                                  float* __restrict__ lse) {
  const int lane = threadIdx.x & 31;
  const int wave = threadIdx.x >> 5;
  const int rowbase = (blockIdx.x * 4 + wave) * 16;

  __shared__ float smax[4][16];

  // ---- Pass A: per-row max, coalesced float4 loads + wave32 shuffle reduce
  for (int r = 0; r < 16; ++r) {
    const float* rowp = logits + (size_t)(rowbase + r) * C_;
    float m = -INFINITY;
#pragma unroll
    for (int i = 0; i < 4; ++i) {
      v4f v = *(const v4f*)(rowp + i * 128 + lane * 4);
      m = fmaxf(m, fmaxf(fmaxf(v.x, v.y), fmaxf(v.z, v.w)));
    }
#pragma unroll
    for (int off = 16; off; off >>= 1)
      m = fmaxf(m, __shfl_xor(m, off, 32));
    if (lane == 0) smax[wave][r] = m;
  }
  __syncthreads();

  // ---- Pass B: WMMA row-sum of exp(x - rowmax) (re-read is L2-resident).
  // A-matrix 16x4 f32 layout: lane l holds row (l&15), cols k0+2h, k0+2h+1
  // (h = l>>4) in its two A VGPRs -> a contiguous float2 load per step.
  const int half = lane >> 4;
  const int myrow = lane & 15;
  const float rm = smax[wave][myrow];
  const float* lanerow = logits + (size_t)(rowbase + myrow) * C_ + 2 * half;

  v2f ones; ones.x = 1.0f; ones.y = 1.0f;   // B = all ones (layout-invariant)
  v8f acc = {};                              // C/D accumulator, starts at 0

  for (int k0 = 0; k0 < C_; k0 += 4) {
    v2f v = *(const v2f*)(lanerow + k0);
    v2f a;
    // args <= 0: fast native exp is safe (no overflow; underflow -> 0)
    a.x = __expf(v.x - rm);
    a.y = __expf(v.y - rm);
    // D = A(16x4) * ones(4x16) + C : every column of D = row sums
    acc = __builtin_amdgcn_wmma_f32_16x16x4_f32(
        /*neg_a=*/false, a, /*neg_b=*/false, ones,
        /*c_mod=*/(short)0, acc, /*reuse_a=*/false, /*reuse_b=*/false);
  }

  // D layout: VGPR j, lanes 0-15 -> M=j ; lanes 16-31 -> M=j+8. Take column 0.
  if (myrow == 0) {
#pragma unroll
    for (int j = 0; j < 8; ++j) {
      int r = j + 8 * half;
      lse[rowbase + r] = smax[wave][r] + logf(acc[j]);
    }
  }
}

// ---------------------------------------------------------------------------
// Kernel 2: CTC forward recursion, one workgroup per batch element.
// Emit gathers for step t+1 are issued before step t's math (latency hiding).
// ---------------------------------------------------------------------------
__device__ __forceinline__ float lae(float x, float y) {
  float m = fmaxf(x, y);
  float n = fminf(x, y);
  return m + log1pf(expf(n - m));
}

__global__ __launch_bounds__(160) void ctc_forward_kernel(
    const float* __restrict__ logits, const float* __restrict__ lse,
    const int* __restrict__ labels, const int* __restrict__ label_length,
    const int* __restrict__ logit_length, float* __restrict__ nll) {
  const int b = blockIdx.x;
  const int tid = threadIdx.x;
  const int blank = C_ - 1;
  const bool act = (tid < S_);

  __shared__ int yext[S_];
  __shared__ float aA[S_], aB[S_];
  __shared__ float fin[2];

  const int lab_len = label_length[b];
  const int log_len = logit_length[b];
  const int endi = 2 * lab_len;   // index of final blank

  if (act) yext[tid] = (tid & 1) ? labels[b * L_ + (tid >> 1)] : blank;
  __syncthreads();

  int y = 0;
  bool skip = false;
  if (act) {
    y = yext[tid];
    skip = (tid >= 2) && (y != blank) && (y != yext[tid - 2]);
  }

  const float* lg = logits + (size_t)b * T_ * C_;
  const float* lz = lse + b * T_;

  float* cur = aA;
  float* nxt = aB;

  // t = 0
  if (act) {
    float e = lg[y] - lz[0];
    cur[tid] = (tid <= 1) ? e : NEGV;
  }
  __syncthreads();
  if (log_len == 1) {
    if (tid == endi)          fin[0] = cur[tid];
    else if (tid == endi - 1) fin[1] = cur[tid];
  }

  // prefetch emit data for t = 1
  float raw_nxt = 0.0f, lse_nxt = 0.0f;
  if (act) raw_nxt = lg[(size_t)1 * C_ + y];
  lse_nxt = lz[1];

  for (int t = 1; t < T_; ++t) {
    float e = raw_nxt - lse_nxt;
    // issue next gather before the dependent alpha math
    if (t + 1 < T_) {
      if (act) raw_nxt = lg[(size_t)(t + 1) * C_ + y];
      lse_nxt = lz[t + 1];
    }
    if (act) {
      float a0 = cur[tid];
      float a1 = (tid >= 1) ? cur[tid - 1] : NEGV;
      float a2 = skip ? cur[tid - 2] : NEGV;
      nxt[tid] = lae(lae(a0, a1), a2) + e;
    }
    __syncthreads();
    if (t == log_len - 1) {                  // uniform condition
      if (tid == endi)          fin[0] = nxt[tid];
      else if (tid == endi - 1) fin[1] = nxt[tid];
    }
    float* tmp = cur; cur = nxt; nxt = tmp;
  }
  __syncthreads();

  if (tid == 0) nll[b] = -lae(fin[0], fin[1]);
}

// ---------------------------------------------------------------------------
// Kernel 3: mean over B
// ---------------------------------------------------------------------------
__global__ __launch_bounds__(256) void mean_kernel(const float* __restrict__ nll,
                                                   float* __restrict__ out) {
  __shared__ float s[256];
  int t = threadIdx.x;
  s[t] = nll[t];
  __syncthreads();
  for (int off = 128; off; off >>= 1) {
    if (t < off) s[t] += s[t + off];
    __syncthreads();
  }
  if (t == 0) out[0] = s[0] / (float)B_;
}

extern "C" void kernel_launch(void* const* d_in, const int* in_sizes, int n_in,
                              void* d_out, int out_size, void* d_ws, size_t ws_size,
                              hipStream_t stream) {
  const float* logits       = (const float*)d_in[0];
  const int*   labels       = (const int*)d_in[1];
  const int*   label_length = (const int*)d_in[2];
  const int*   logit_length = (const int*)d_in[3];

  float* lse = (float*)d_ws;            // B*T floats
  float* nll = lse + (size_t)B_ * T_;   // B floats

  // 65536 rows, 64 rows per block (4 waves x 16 rows)
  lse_kernel<<<(B_ * T_) / 64, 128, 0, stream>>>(logits, lse);
  ctc_forward_kernel<<<B_, 160, 0, stream>>>(logits, lse, labels, label_length,
                                             logit_length, nll);
  mean_kernel<<<1, 256, 0, stream>>>(nll, (float*)d_out);
}